// EnhancedTGNN_55611236549196
// MI455X (gfx1250) — compile-verified
//
#include <hip/hip_runtime.h>
#include <math.h>

typedef float v2f __attribute__((ext_vector_type(2)));
typedef float v8f __attribute__((ext_vector_type(8)));

#define CDIV(a, b) (((a) + (b) - 1) / (b))

// ---------------------------------------------------------------------------
// WMMA f32 GEMM:  Out[M,Nout] = A[M,K] @ W[Nout,K]^T  (+bias) (+BN) (+ReLU)
// One wave computes one 16x16 output tile via V_WMMA_F32_16X16X4_F32.
// ---------------------------------------------------------------------------
__global__ __launch_bounds__(256)
void gemm_wmma_f32_kernel(const float* __restrict__ A,   // [M,K] row-major
                          const float* __restrict__ W,   // [Nout,K] row-major (B[k][n]=W[n][k])
                          float* __restrict__ Out,       // [M,Nout]
                          const float* __restrict__ bias, // [Nout] or null
                          const float* __restrict__ bng, const float* __restrict__ bnb,
                          const float* __restrict__ bnrm, const float* __restrict__ bnrv,
                          int M, int K, int Nout, int relu)
{
    const int wave = (int)((blockIdx.x * blockDim.x + threadIdx.x) >> 5);
    const int lane = (int)(threadIdx.x & 31);
    const int tn = Nout >> 4;
    const int tiles = (M >> 4) * tn;
    if (wave >= tiles) return;                 // wave-uniform: EXEC stays all-1s

    const int tile_n = wave % tn;
    const int tile_m = wave / tn;
    const int mrow = (tile_m << 4) + (lane & 15);   // A row served by this lane
    const int ncol = (tile_n << 4) + (lane & 15);   // B col served by this lane
    const int khalf = (lane >> 4) << 1;             // 0 (lanes 0-15) or 2 (lanes 16-31)

    const float* __restrict__ aptr = A + (size_t)mrow * K + khalf;
    const float* __restrict__ wptr = W + (size_t)ncol * K + khalf;

    v8f acc = {};
    for (int k = 0; k < K; k += 4) {
        v2f a; a.x = aptr[k]; a.y = aptr[k + 1];
        v2f b; b.x = wptr[k]; b.y = wptr[k + 1];
        acc = __builtin_amdgcn_wmma_f32_16x16x4_f32(
            /*neg_a=*/false, a, /*neg_b=*/false, b,
            /*c_mod=*/(short)0, acc, /*reuse_a=*/false, /*reuse_b=*/false);
    }

    // per-column epilogue: val = acc*scale + shift ; optional relu
    float scale = 1.0f;
    float shift = bias ? bias[ncol] : 0.0f;
    if (bng) {
        const float inv = rsqrtf(bnrv[ncol] + 1e-5f);
        scale = bng[ncol] * inv;
        shift = (shift - bnrm[ncol]) * scale + bnb[ncol];
    }

    const int rbase = (tile_m << 4) + ((lane >> 4) << 3);  // +v gives row of acc[v]
    float* __restrict__ optr = Out + (size_t)rbase * Nout + ncol;
    #pragma unroll
    for (int v = 0; v < 8; ++v) {
        float val = acc[v] * scale + shift;
        if (relu) val = fmaxf(val, 0.0f);
        optr[(size_t)v * Nout] = val;
    }
}

// ---------------------------------------------------------------------------
// Utility kernels
// ---------------------------------------------------------------------------
__global__ void fill_f32_kernel(float* __restrict__ p, float v, long long n) {
    long long t = (long long)blockIdx.x * blockDim.x + threadIdx.x;
    if (t < n) p[t] = v;
}

__global__ void reduce_sum_kernel(const float* __restrict__ w, int n,
                                  float* __restrict__ partials) {
    __shared__ float sm[256];
    float s = 0.0f;
    for (int i = blockIdx.x * blockDim.x + threadIdx.x; i < n; i += gridDim.x * blockDim.x)
        s += w[i];
    sm[threadIdx.x] = s;
    __syncthreads();
    for (int st = 128; st > 0; st >>= 1) {
        if ((int)threadIdx.x < st) sm[threadIdx.x] += sm[threadIdx.x + st];
        __syncthreads();
    }
    if (threadIdx.x == 0) partials[blockIdx.x] = sm[0];
}

__global__ void finalize_mean_kernel(const float* __restrict__ partials, int nb,
                                     float invn, float* __restrict__ out) {
    __shared__ float sm[256];
    float s = ((int)threadIdx.x < nb) ? partials[threadIdx.x] : 0.0f;
    sm[threadIdx.x] = s;
    __syncthreads();
    for (int st = 128; st > 0; st >>= 1) {
        if ((int)threadIdx.x < st) sm[threadIdx.x] += sm[threadIdx.x + st];
        __syncthreads();
    }
    if (threadIdx.x == 0) *out = sm[0] * invn;
}

// GRUCell gates: h = (1-z)*tanh(i_n + r*h_n) + z*h_prev
__global__ void gru_gates_kernel(const float* __restrict__ gi,   // [N,192]
                                 const float* __restrict__ gh,   // [N,192]
                                 const float* __restrict__ hprev,// [N,64]
                                 float* __restrict__ hout, int N) {
    int t = blockIdx.x * blockDim.x + threadIdx.x;
    if (t >= N * 64) return;
    int n = t >> 6, c = t & 63;
    const float* gin = gi + (size_t)n * 192;
    const float* ghn = gh + (size_t)n * 192;
    float rg = 1.0f / (1.0f + expf(-(gin[c] + ghn[c])));
    float zg = 1.0f / (1.0f + expf(-(gin[64 + c] + ghn[64 + c])));
    float ng = tanhf(gin[128 + c] + rg * ghn[128 + c]);
    hout[t] = (1.0f - zg) * ng + zg * hprev[t];
}

// ---------------------------------------------------------------------------
// GATv2 edge phase. EA = E + N edges (self-loops appended, ea = mean weight).
// ---------------------------------------------------------------------------
// Pass 1: per-edge attention logits + segment-max (ordered-uint atomicMax).
// One wave per edge; lane = head*8 + i, each lane covers 8 channels.
__global__ __launch_bounds__(256)
void gat_logits_kernel(const float* __restrict__ xl, const float* __restrict__ xr,
                       const int* __restrict__ edge_index,
                       const float* __restrict__ edge_weight,
                       const float* __restrict__ meanw,
                       const float* __restrict__ we,   // [256]
                       const float* __restrict__ att,  // [H=4,64] flat
                       float* __restrict__ alog,       // [EA,4]
                       unsigned int* __restrict__ amax,// [N,4] (init 0 == mapped -inf)
                       int E, int N)
{
    const int wid = (int)((blockIdx.x * blockDim.x + threadIdx.x) >> 5);
    const int lane = (int)(threadIdx.x & 31);
    const int EA = E + N;
    if (wid >= EA) return;

    int src, tgt; float ea;
    if (wid < E) { src = edge_index[wid]; tgt = edge_index[E + wid]; ea = edge_weight[wid]; }
    else         { src = tgt = wid - E;   ea = *meanw; }

    const int h = lane >> 3, i = lane & 7;
    const int base = h * 64 + i * 8;
    const float4* pl = (const float4*)(xl + (size_t)src * 256 + base);
    const float4* pr = (const float4*)(xr + (size_t)tgt * 256 + base);
    const float4* pw = (const float4*)(we + base);
    const float4* pa = (const float4*)(att + base);

    float p = 0.0f;
    #pragma unroll
    for (int q = 0; q < 2; ++q) {
        float4 l = pl[q], r = pr[q], w4 = pw[q], a4 = pa[q];
        float m0 = l.x + r.x + ea * w4.x; m0 = (m0 > 0.f) ? m0 : 0.2f * m0;
        float m1 = l.y + r.y + ea * w4.y; m1 = (m1 > 0.f) ? m1 : 0.2f * m1;
        float m2 = l.z + r.z + ea * w4.z; m2 = (m2 > 0.f) ? m2 : 0.2f * m2;
        float m3 = l.w + r.w + ea * w4.w; m3 = (m3 > 0.f) ? m3 : 0.2f * m3;
        p += m0 * a4.x + m1 * a4.y + m2 * a4.z + m3 * a4.w;
    }
    // reduce across the 8-lane head group (aligned, so xor 1/2/4 stays inside)
    p += __shfl_xor(p, 1);
    p += __shfl_xor(p, 2);
    p += __shfl_xor(p, 4);

    if (i == 0) {
        alog[(size_t)wid * 4 + h] = p;
        unsigned int u = __float_as_uint(p);
        u = (u & 0x80000000u) ? ~u : (u | 0x80000000u);   // order-preserving map
        atomicMax(&amax[(size_t)tgt * 4 + h], u);
    }
}

// Pass 2: ex = exp(a - amax[tgt]) (stored in-place) + segment-sum into den.
__global__ void gat_softmax_kernel(float* __restrict__ alog,
                                   const unsigned int* __restrict__ amax,
                                   float* __restrict__ den,
                                   const int* __restrict__ edge_index, int E, int N) {
    const int EA = E + N;
    int t = blockIdx.x * blockDim.x + threadIdx.x;
    if (t >= EA * 4) return;
    int e = t >> 2, h = t & 3;
    int tgt = (e < E) ? edge_index[E + e] : (e - E);
    unsigned int u = amax[(size_t)tgt * 4 + h];
    unsigned int ob = (u & 0x80000000u) ? (u & 0x7FFFFFFFu) : ~u;  // inverse map
    float ex = expf(alog[t] - __uint_as_float(ob));
    alog[t] = ex;
    atomicAdd(&den[(size_t)tgt * 4 + h], ex);
}

// Pass 3: agg[tgt] += alpha * xl[src]. One wave per edge, 8 ch per lane.
__global__ __launch_bounds__(256)
void gat_aggregate_kernel(const float* __restrict__ xl, const float* __restrict__ ex,
                          const float* __restrict__ den,
                          const int* __restrict__ edge_index,
                          float* __restrict__ agg, int E, int N) {
    const int wid = (int)((blockIdx.x * blockDim.x + threadIdx.x) >> 5);
    const int lane = (int)(threadIdx.x & 31);
    const int EA = E + N;
    if (wid >= EA) return;
    int src, tgt;
    if (wid < E) { src = edge_index[wid]; tgt = edge_index[E + wid]; }
    else         { src = tgt = wid - E; }
    const int h = lane >> 3, i = lane & 7;
    const float alpha = ex[(size_t)wid * 4 + h] / (den[(size_t)tgt * 4 + h] + 1e-16f);
    const int base = h * 64 + i * 8;
    const float* pl = xl + (size_t)src * 256 + base;
    float* pd = agg + (size_t)tgt * 256 + base;
    #pragma unroll
    for (int j = 0; j < 8; ++j) atomicAdd(&pd[j], alpha * pl[j]);
}

// Pass 4: head-mean + layer bias + BN + ReLU + residual.
__global__ void gat_epilogue_kernel(const float* __restrict__ agg,   // [N,4,64]
                                    const float* __restrict__ gbias, // [64]
                                    const float* __restrict__ bng, const float* __restrict__ bnb,
                                    const float* __restrict__ bnrm, const float* __restrict__ bnrv,
                                    const float* __restrict__ hin, float* __restrict__ hout, int N) {
    int t = blockIdx.x * blockDim.x + threadIdx.x;
    if (t >= N * 64) return;
    int n = t >> 6, c = t & 63;
    const float* p = agg + (size_t)n * 256 + c;
    float m = 0.25f * (p[0] + p[64] + p[128] + p[192]) + gbias[c];
    float v = (m - bnrm[c]) * bng[c] * rsqrtf(bnrv[c] + 1e-5f) + bnb[c];
    hout[t] = fmaxf(v, 0.0f) + hin[t];
}

// L2 row normalize: one wave per node, 2 channels per lane.
__global__ __launch_bounds__(256)
void l2norm_kernel(const float* __restrict__ h, float* __restrict__ out, int N) {
    const int wid = (int)((blockIdx.x * blockDim.x + threadIdx.x) >> 5);
    const int lane = (int)(threadIdx.x & 31);
    if (wid >= N) return;
    float a = h[(size_t)wid * 64 + lane];
    float b = h[(size_t)wid * 64 + 32 + lane];
    float ss = a * a + b * b;
    #pragma unroll
    for (int m = 1; m < 32; m <<= 1) ss += __shfl_xor(ss, m);
    float inv = 1.0f / fmaxf(sqrtf(ss), 1e-12f);
    out[(size_t)wid * 64 + lane] = a * inv;
    out[(size_t)wid * 64 + 32 + lane] = b * inv;
}

// Pair scores: s^T W d reduced to dot(s, V[d]) with V = h_out @ W^T (WMMA GEMM).
__global__ __launch_bounds__(256)
void pair_score_kernel(const float* __restrict__ hout, const float* __restrict__ V,
                       const int* __restrict__ pair_index,
                       const float* __restrict__ bilb,
                       float* __restrict__ scores, int P) {
    const int wid = (int)((blockIdx.x * blockDim.x + threadIdx.x) >> 5);
    const int lane = (int)(threadIdx.x & 31);
    if (wid >= P) return;
    const int s = pair_index[wid], d = pair_index[P + wid];
    const float* ps = hout + (size_t)s * 64;
    const float* pd = hout + (size_t)d * 64;
    const float* pv = V + (size_t)d * 64;
    float s0 = ps[lane], s1 = ps[32 + lane];
    float d0 = pd[lane], d1 = pd[32 + lane];
    float v0 = pv[lane], v1 = pv[32 + lane];
    float sv = s0 * v0 + s1 * v1;
    float sd = s0 * d0 + s1 * d1;
    float ss = s0 * s0 + s1 * s1;
    float dd = d0 * d0 + d1 * d1;
    #pragma unroll
    for (int m = 1; m < 32; m <<= 1) {
        sv += __shfl_xor(sv, m);
        sd += __shfl_xor(sd, m);
        ss += __shfl_xor(ss, m);
        dd += __shfl_xor(dd, m);
    }
    if (lane == 0) {
        float score = sv + bilb[0];
        float cosv = sd / fmaxf(sqrtf(ss) * sqrtf(dd), 1e-8f);
        scores[wid] = score + cosv;
    }
}

// ---------------------------------------------------------------------------
// Host-side orchestration
// ---------------------------------------------------------------------------
extern "C" void kernel_launch(void* const* d_in, const int* in_sizes, int n_in,
                              void* d_out, int out_size, void* d_ws, size_t ws_size,
                              hipStream_t stream) {
    const float* x          = (const float*)d_in[0];
    const int*   edge_index = (const int*)d_in[1];
    const float* edge_w     = (const float*)d_in[2];
    const int*   pair_index = (const int*)d_in[3];
    const float* hidden     = (const float*)d_in[4];
    const float* enc_w      = (const float*)d_in[5];
    const float* enc_b      = (const float*)d_in[6];
    const float* bn0g = (const float*)d_in[7],  *bn0b = (const float*)d_in[8];
    const float* bn0rm = (const float*)d_in[9], *bn0rv = (const float*)d_in[10];
    const float* gru_wih = (const float*)d_in[11], *gru_whh = (const float*)d_in[12];
    const float* gru_bih = (const float*)d_in[13], *gru_bhh = (const float*)d_in[14];
    const float* g1_wl = (const float*)d_in[15], *g1_bl = (const float*)d_in[16];
    const float* g1_wr = (const float*)d_in[17], *g1_br = (const float*)d_in[18];
    const float* g1_att = (const float*)d_in[19], *g1_we = (const float*)d_in[20];
    const float* g1_bias = (const float*)d_in[21];
    const float* bn1g = (const float*)d_in[22], *bn1b = (const float*)d_in[23];
    const float* bn1rm = (const float*)d_in[24], *bn1rv = (const float*)d_in[25];
    const float* g2_wl = (const float*)d_in[26], *g2_bl = (const float*)d_in[27];
    const float* g2_wr = (const float*)d_in[28], *g2_br = (const float*)d_in[29];
    const float* g2_att = (const float*)d_in[30], *g2_we = (const float*)d_in[31];
    const float* g2_bias = (const float*)d_in[32];
    const float* bn2g = (const float*)d_in[33], *bn2b = (const float*)d_in[34];
    const float* bn2rm = (const float*)d_in[35], *bn2rv = (const float*)d_in[36];
    const float* bil_w = (const float*)d_in[37];
    const float* bil_b = (const float*)d_in[38];

    const int N = in_sizes[4] / 64;
    const int E = in_sizes[2];
    const int P = in_sizes[3] / 2;
    const int EA = E + N;

    // workspace layout (floats)
    float* ws = (float*)d_ws;
    const size_t NF = (size_t)N * 64;
    float* hA   = ws;                         // [N,64]
    float* hB   = ws + NF;                    // [N,64]   (also h_enc)
    float* xl   = ws + 2 * NF;                // [N,256]  (also gi [N,192], also V [N,64])
    float* xr   = ws + 2 * NF + (size_t)N * 256;  // [N,256] (also gh, also agg)
    float* alog = ws + 2 * NF + 2 * (size_t)N * 256;            // [EA,4]
    float* amax = alog + (size_t)EA * 4;      // [N,4] (ordered-uint segment max)
    float* den  = amax + (size_t)N * 4;       // [N,4]
    float* part = den + (size_t)N * 4;        // [256]
    float* meanw = part + 256;                // [1]

    float* out    = (float*)d_out;
    float* scores = out;        // [P]
    float* hout   = out + P;    // [N,64]

    const int TB = 256;

    auto gemm = [&](const float* A, const float* W, float* O, const float* bias,
                    const float* g, const float* b, const float* rm, const float* rv,
                    int M, int K, int Nout, int relu) {
        int tiles = (M / 16) * (Nout / 16);
        int blocks = CDIV(tiles, 8);  // 8 waves / block
        gemm_wmma_f32_kernel<<<blocks, TB, 0, stream>>>(A, W, O, bias, g, b, rm, rv,
                                                        M, K, Nout, relu);
    };

    // 0) mean edge weight (deterministic 2-pass reduction)
    reduce_sum_kernel<<<256, TB, 0, stream>>>(edge_w, E, part);
    finalize_mean_kernel<<<1, TB, 0, stream>>>(part, 256, 1.0f / (float)E, meanw);

    // 1) encoder: h_enc = relu(bn0(x @ enc_w^T + enc_b))  -> hB
    gemm(x, enc_w, hB, enc_b, bn0g, bn0b, bn0rm, bn0rv, N, 128, 64, 1);

    // 2) GRU: gi = h_enc@wih^T+bih -> xl ; gh = hidden@whh^T+bhh -> xr ; gates -> hA
    gemm(hB, gru_wih, xl, gru_bih, nullptr, nullptr, nullptr, nullptr, N, 64, 192, 0);
    gemm(hidden, gru_whh, xr, gru_bhh, nullptr, nullptr, nullptr, nullptr, N, 64, 192, 0);
    gru_gates_kernel<<<CDIV(N * 64, TB), TB, 0, stream>>>(xl, xr, hidden, hA, N);

    // 3) two GATv2 layers
    auto run_gat = [&](const float* hin, float* hnext,
                       const float* wl, const float* bl, const float* wr, const float* br,
                       const float* att, const float* wE, const float* gbias,
                       const float* bg, const float* bb, const float* brm, const float* brv) {
        gemm(hin, wl, xl, bl, nullptr, nullptr, nullptr, nullptr, N, 64, 256, 0);
        gemm(hin, wr, xr, br, nullptr, nullptr, nullptr, nullptr, N, 64, 256, 0);
        fill_f32_kernel<<<CDIV(N * 4, TB), TB, 0, stream>>>(amax, 0.0f, (long long)N * 4);
        gat_logits_kernel<<<CDIV(EA, 8), TB, 0, stream>>>(
            xl, xr, edge_index, edge_w, meanw, wE, att, alog, (unsigned int*)amax, E, N);
        // xr is dead now -> reuse its slab as agg
        fill_f32_kernel<<<CDIV(N * 256, TB), TB, 0, stream>>>(xr, 0.0f, (long long)N * 256);
        fill_f32_kernel<<<CDIV(N * 4, TB), TB, 0, stream>>>(den, 0.0f, (long long)N * 4);
        gat_softmax_kernel<<<CDIV(EA * 4, TB), TB, 0, stream>>>(
            alog, (const unsigned int*)amax, den, edge_index, E, N);
        gat_aggregate_kernel<<<CDIV(EA, 8), TB, 0, stream>>>(xl, alog, den, edge_index, xr, E, N);
        gat_epilogue_kernel<<<CDIV(N * 64, TB), TB, 0, stream>>>(
            xr, gbias, bg, bb, brm, brv, hin, hnext, N);
    };

    run_gat(hA, hB, g1_wl, g1_bl, g1_wr, g1_br, g1_att, g1_we, g1_bias,
            bn1g, bn1b, bn1rm, bn1rv);
    run_gat(hB, hA, g2_wl, g2_bl, g2_wr, g2_br, g2_att, g2_we, g2_bias,
            bn2g, bn2b, bn2rm, bn2rv);

    // 4) L2 normalize -> h_out (second output, written straight into d_out)
    l2norm_kernel<<<CDIV(N, 8), TB, 0, stream>>>(hA, hout, N);

    // 5) bilinear: V = h_out @ bil_w^T  (WMMA), then per-pair dots
    gemm(hout, bil_w, xl, nullptr, nullptr, nullptr, nullptr, nullptr, N, 64, 64, 0);
    pair_score_kernel<<<CDIV(P, 8), TB, 0, stream>>>(hout, xl, pair_index, bil_b, scores, P);
}